// FCOSDetect_91061896610318
// MI455X (gfx1250) — compile-verified
//
#include <hip/hip_runtime.h>
#include <stdint.h>

typedef __attribute__((ext_vector_type(2))) float v2f;
typedef __attribute__((ext_vector_type(8))) float v8f;
typedef __attribute__((ext_vector_type(4))) unsigned int u32x4;
typedef __attribute__((ext_vector_type(8))) int i32x8;
typedef __attribute__((ext_vector_type(4))) int i32x4;

#define NSORT 32768
#define KPAD  1024
#define KTOP  1000
#define SCORE_THR 0.05f
#define IOU_THR   0.5f
#define XLIM 1023.0f
#define YLIM 799.0f

// ---------------------------------------------------------------------------
// Kernel A: decode scores, build sortable keys (score bits desc, idx asc ties)
// ---------------------------------------------------------------------------
__global__ __launch_bounds__(256) void fcos_decode(
    const float* __restrict__ cls, const float* __restrict__ ctr,
    unsigned long long* __restrict__ keys, int n, int B)
{
  int t = blockIdx.x * blockDim.x + threadIdx.x;
  int b = t >> 15;              // 32768 slots per batch
  int i = t & (NSORT - 1);
  if (b >= B) return;
  unsigned long long key = 0ull;
  if (i < n) {
    size_t base = (size_t)b * (size_t)n + (size_t)i;
    const float* cp = cls + base * 4;
    float best = fmaxf(fmaxf(cp[0], cp[1]), fmaxf(cp[2], cp[3]));
    float ms = 1.0f / (1.0f + expf(-best));        // sigmoid(max) == max(sigmoid)
    float cs = 1.0f / (1.0f + expf(-ctr[base]));
    float score = sqrtf(ms * cs);                  // score >= 0 -> bits monotone
    key = ((unsigned long long)__float_as_uint(score) << 32) |
          (unsigned long long)(0xFFFFFFFFu - (unsigned)i);
  }
  keys[(size_t)b * NSORT + i] = key;
}

// ---------------------------------------------------------------------------
// Kernel B: per-batch block (1024 thr = 32 waves): sort + topk + NMS + output
// ---------------------------------------------------------------------------
__global__ __launch_bounds__(1024) void fcos_sort_nms(
    const float* __restrict__ cls, const float* __restrict__ reg,
    const float* __restrict__ coords, const float* __restrict__ pstr,
    unsigned long long* __restrict__ keys_all,
    float* __restrict__ out, int n)
{
  const int tid = threadIdx.x;
  const int b   = blockIdx.x;
  const int B   = gridDim.x;
  unsigned long long* keys = keys_all + (size_t)b * NSORT;

  // 128KB multi-purpose LDS: phase1 = staged keys (8KB) + reduction scratch,
  // phase4+ = 1024x1024 suppression bit matrix (u16 halfwords, 64 per row).
  __shared__ unsigned long long smemU[16384];               // 131072 B
  __shared__ float sx1[KPAD], sy1[KPAD], sx2[KPAD], sy2[KPAD], sarea[KPAD];
  __shared__ unsigned skeep[KPAD];

  // ---- Phase 0: bitonic sort, descending by key (global ws, L2 resident) --
  for (unsigned k = 2; k <= NSORT; k <<= 1) {
    for (unsigned j = k >> 1; j > 0; j >>= 1) {
      for (unsigned i = tid; i < NSORT; i += 1024) {
        unsigned ixj = i ^ j;
        if (ixj > i) {
          unsigned long long a = keys[i], c = keys[ixj];
          bool up = (i & k) == 0;
          if (up ? (a < c) : (a > c)) { keys[i] = c; keys[ixj] = a; }
        }
      }
      __syncthreads();
    }
  }
  __threadfence();              // make sorted keys visible to the TDM engine
  __syncthreads();

  // ---- Phase 1: TDM async stage of top-1024 keys (8KB) into LDS ----------
  unsigned long long* kb = smemU;
#if __has_builtin(__builtin_amdgcn_tensor_load_to_lds) && __has_builtin(__builtin_amdgcn_s_wait_tensorcnt)
  if (tid < 32) {   // single wave issues the TDM op (EXEC ignored by TDM)
    unsigned lds_off = (unsigned)(unsigned long long)(void*)kb; // low32 of LDS generic addr
    unsigned long long ga = (unsigned long long)keys;
    // D# group0: count=1 | lds_addr | global_addr[56:0] | type=2
    u32x4 g0 = { 1u, lds_off, (unsigned)ga,
                 ((unsigned)((ga >> 32) & 0x01FFFFFFull)) | 0x80000000u };
    // D# group1: data_size=3 (8B); tensor_dim0=32768; tensor_dim1=1;
    //            tile_dim0=1024; tile_dim1=1; dim0_stride=32768
    i32x8 g1 = { (int)(3u << 16),
                 (int)(((unsigned)NSORT & 0xFFFFu) << 16),
                 (int)(1u << 16),
                 (int)(1024u << 16),
                 1,
                 (int)NSORT,
                 0, 0 };
    i32x4 gz = { 0, 0, 0, 0 };
  #if defined(__clang_major__) && (__clang_major__ >= 23)
    i32x8 gz8 = { 0, 0, 0, 0, 0, 0, 0, 0 };
    __builtin_amdgcn_tensor_load_to_lds(g0, g1, gz, gz, gz8, 0);
  #else
    __builtin_amdgcn_tensor_load_to_lds(g0, g1, gz, gz, 0);
  #endif
    __builtin_amdgcn_s_wait_tensorcnt(0);
  }
#else
  kb[tid] = keys[tid];
#endif
  __syncthreads();

  // ---- Phase 1b: unpack my top-k entry; recompute label + clipped box ----
  unsigned long long mykey = kb[tid];
  float    myscore = __uint_as_float((unsigned)(mykey >> 32));
  unsigned myidx   = 0xFFFFFFFFu - (unsigned)(mykey & 0xFFFFFFFFull);
  bool valid = (tid < KTOP) && (myscore >= SCORE_THR);
  float ox1 = 0.f, oy1 = 0.f, ox2 = 0.f, oy2 = 0.f, lblf = 0.f;
  if (tid < KTOP) {
    size_t base = (size_t)b * (size_t)n + (size_t)myidx;
    const float* cp = cls + base * 4;
    float c0 = cp[0], c1 = cp[1], c2 = cp[2], c3 = cp[3];
    int lab = 0; float best = c0;
    if (c1 > best) { best = c1; lab = 1; }
    if (c2 > best) { best = c2; lab = 2; }
    if (c3 > best) { best = c3; lab = 3; }
    lblf = (float)(lab + 1);                    // argmax (first max) + 1
    float cx = coords[(size_t)myidx * 2 + 0];
    float cy = coords[(size_t)myidx * 2 + 1];
    float s  = pstr[myidx];
    const float* rp = reg + base * 4;
    ox1 = fminf(fmaxf(cx - rp[0] * s, 0.f), XLIM);
    oy1 = fminf(fmaxf(cy - rp[1] * s, 0.f), YLIM);
    ox2 = fminf(fmaxf(cx + rp[2] * s, 0.f), XLIM);
    oy2 = fminf(fmaxf(cy + rp[3] * s, 0.f), YLIM);
  }

  // ---- Phase 2: mx/mn over valid boxes (block reduction, scratch in smemU)
  float* red = (float*)(smemU + 1024);          // byte offset 8192, disjoint from kb
  float lmax = valid ? fmaxf(fmaxf(ox1, oy1), fmaxf(ox2, oy2)) : -1e9f;
  float lmin = valid ? fminf(fminf(ox1, oy1), fminf(ox2, oy2)) :  1e9f;
  red[tid] = lmax; __syncthreads();
  for (int s2 = 512; s2 > 0; s2 >>= 1) {
    if (tid < s2) red[tid] = fmaxf(red[tid], red[tid + s2]);
    __syncthreads();
  }
  float mx = red[0]; __syncthreads();
  red[tid] = lmin; __syncthreads();
  for (int s2 = 512; s2 > 0; s2 >>= 1) {
    if (tid < s2) red[tid] = fminf(red[tid], red[tid + s2]);
    __syncthreads();
  }
  float mn = red[0]; __syncthreads();

  // ---- Phase 3: class-offset boxes + areas + keep flags into LDS ---------
  float off = lblf * (mx - mn + 1.0f);
  float bx1 = ox1 + off, by1 = oy1 + off, bx2 = ox2 + off, by2 = oy2 + off;
  if (tid >= KTOP) { bx1 = by1 = bx2 = by2 = 0.f; }
  sx1[tid] = bx1; sy1[tid] = by1; sx2[tid] = bx2; sy2[tid] = by2;
  sarea[tid] = (bx2 - bx1) * (by2 - by1);
  skeep[tid] = valid ? 1u : 0u;
  __syncthreads();

  // ---- Phase 4: suppression bit matrix, one 16x16 tile per wave-iteration.
  // areaSum tile (area_m + area_n) via two V_WMMA_F32_16X16X4_F32; the
  // construction is invariant to the A/B K-slot mapping (uniform per k).
  unsigned short* sup = (unsigned short*)smemU;  // sup[m*64 + J] = 16 bits
  const int wave = tid >> 5;
  const int lane = tid & 31;
  const int ln15 = lane & 15;
  const int mo   = (lane >> 4) << 3;             // C/D layout: lanes>=16 -> M+8
  for (int t = wave; t < 64 * 64; t += 32) {
    const int I = t >> 6, J = t & 63;
    const int mBase = I << 4, nBase = J << 4;
    float am = sarea[mBase + ln15];
    float an = sarea[nBase + ln15];
    float dsum[8];
#if __has_builtin(__builtin_amdgcn_wmma_f32_16x16x4_f32)
    v2f a1 = { 0.25f * am, 0.25f * am };
    v2f b1 = { 1.0f, 1.0f };
    v8f cz = {};
    v8f d1 = __builtin_amdgcn_wmma_f32_16x16x4_f32(
        false, a1, false, b1, (short)0, cz, false, false);   // D1[m][n]=area_m
    v2f a2 = { 0.25f, 0.25f };
    v2f b2 = { an, an };
    v8f d2 = __builtin_amdgcn_wmma_f32_16x16x4_f32(
        false, a2, false, b2, (short)0, d1, false, false);   // +=area_n
    #pragma unroll
    for (int r = 0; r < 8; ++r) dsum[r] = d2[r];
#else
    #pragma unroll
    for (int r = 0; r < 8; ++r) dsum[r] = sarea[mBase + r + mo] + an;
#endif
    float nx1 = sx1[nBase + ln15], ny1 = sy1[nBase + ln15];
    float nx2 = sx2[nBase + ln15], ny2 = sy2[nBase + ln15];
    #pragma unroll
    for (int r = 0; r < 8; ++r) {
      int m = mBase + r + mo;
      float ix1 = fmaxf(sx1[m], nx1);
      float iy1 = fmaxf(sy1[m], ny1);
      float ix2 = fminf(sx2[m], nx2);
      float iy2 = fminf(sy2[m], ny2);
      float inter = fmaxf(ix2 - ix1, 0.f) * fmaxf(iy2 - iy1, 0.f);
      float uni   = dsum[r] - inter;
      float iou   = inter / fmaxf(uni, 1e-9f);
      unsigned mask = (unsigned)__ballot(iou > IOU_THR ? 1 : 0);
      if (lane == 0) {
        sup[(mBase + r)     * 64 + J] = (unsigned short)(mask & 0xFFFFu);
        sup[(mBase + r + 8) * 64 + J] = (unsigned short)(mask >> 16);
      }
    }
  }
  __syncthreads();

  // ---- Phase 5: greedy suppression, single wave over packed keep bits ----
  if (tid < 32) {
    const unsigned* sup32 = (const unsigned*)smemU;  // sup32[i*32 + word]
    int l = tid;
    unsigned keepw = 0;
    #pragma unroll 4
    for (int q = 0; q < 32; ++q) keepw |= (skeep[l * 32 + q] & 1u) << q;
    for (int i = 0; i < KTOP; ++i) {
      int wi = i >> 5;
      unsigned srcw = __shfl(keepw, wi, 32);
      if ((srcw >> (i & 31)) & 1u) {               // cur = keep[i]
        unsigned supw = sup32[i * 32 + l];
        unsigned jmask = (l > wi) ? 0xFFFFFFFFu
                       : ((l == wi) ? (0xFFFFFFFEu << (i & 31)) : 0u);
        keepw &= ~(supw & jmask);                  // keep &= ~(sup & (idx>i))
      }
    }
    #pragma unroll 4
    for (int q = 0; q < 32; ++q) skeep[l * 32 + q] = (keepw >> q) & 1u;
  }
  __syncthreads();

  // ---- Phase 6: outputs (scores | labels-as-float | boxes) ---------------
  if (tid < KTOP) {
    size_t o = (size_t)b * KTOP + tid;
    bool kp = skeep[tid] != 0u;
    out[o] = kp ? myscore : 0.f;
    out[(size_t)B * KTOP + o] = kp ? lblf : 0.f;
    float* ob = out + (size_t)B * KTOP * 2 + o * 4;
    ob[0] = kp ? ox1 : 0.f;
    ob[1] = kp ? oy1 : 0.f;
    ob[2] = kp ? ox2 : 0.f;
    ob[3] = kp ? oy2 : 0.f;
  }
}

// ---------------------------------------------------------------------------
extern "C" void kernel_launch(void* const* d_in, const int* in_sizes, int n_in,
                              void* d_out, int out_size, void* d_ws, size_t ws_size,
                              hipStream_t stream) {
  const float* cls    = (const float*)d_in[0];
  const float* ctr    = (const float*)d_in[1];
  const float* reg    = (const float*)d_in[2];
  const float* coords = (const float*)d_in[3];
  const float* pstr   = (const float*)d_in[4];
  int n = in_sizes[4];                       // number of anchor points
  int B = (n > 0) ? (in_sizes[1] / n) : 0;   // ctr_logits is B*n
  unsigned long long* keys = (unsigned long long*)d_ws;   // B*32768*8 bytes

  int totalA = B * NSORT;
  fcos_decode<<<(totalA + 255) / 256, 256, 0, stream>>>(cls, ctr, keys, n, B);
  fcos_sort_nms<<<B, 1024, 0, stream>>>(cls, reg, coords, pstr, keys,
                                        (float*)d_out, n);
}